// Loss_Labels_19825569038545
// MI455X (gfx1250) — compile-verified
//
#include <hip/hip_runtime.h>

// ---------------------------------------------------------------------------
// out = (1/(2B)) * sum_i ( d_i + 2*ln(1+exp(-d_i)) ),  d_i = |syn_i - ant_i|
// Bandwidth-bound streaming reduction: 128 MiB in, 1 float out.
// Pass 1: grid-stride, b128 NT loads, per-wave sum via V_WMMA_F32_16X16X4_F32
//         (A = 64 lane accumulators as 16x4 f32, B = ones) -> d_ws partials.
// Pass 2: single block reduces partials, scales by 1/(2B), writes d_out.
// ---------------------------------------------------------------------------

typedef __attribute__((ext_vector_type(4))) float v4f;
typedef __attribute__((ext_vector_type(2))) float v2f;
typedef __attribute__((ext_vector_type(8))) float v8f;

#define NBLK 1024
#define NTHR 256
#define NWAVE (NTHR / 32)
#define NPART (NBLK * NWAVE)

__device__ __forceinline__ float nll_term(float s, float a) {
    // d + 2*ln(1+e^{-d}) computed in base 2 with native transcendentals:
    // t = exp2(-d*log2(e)); u = log2(1+t); result = d + (2*ln2)*u
    float d = __builtin_fabsf(s - a);
    float t = __builtin_amdgcn_exp2f(-1.44269504088896340736f * d);  // v_exp_f32
    float u = __builtin_amdgcn_logf(1.0f + t);                       // v_log_f32 (log2)
    return __builtin_fmaf(1.38629436111989061883f, u, d);
}

__global__ __launch_bounds__(NTHR) void loss_partial_kernel(
    const float* __restrict__ syn, const float* __restrict__ ant,
    float* __restrict__ ws, int n) {
    const int tid    = blockIdx.x * NTHR + threadIdx.x;
    const int stride = NBLK * NTHR;
    const int n4     = n >> 2;

    const v4f* __restrict__ s4 = (const v4f*)syn;
    const v4f* __restrict__ a4 = (const v4f*)ant;

    float acc0 = 0.0f, acc1 = 0.0f;   // two independent chains for ILP
    for (int i = tid; i < n4; i += stride) {
        v4f s = __builtin_nontemporal_load(&s4[i]);   // global_load_b128 th:NT
        v4f a = __builtin_nontemporal_load(&a4[i]);
        acc0 += nll_term(s.x, a.x) + nll_term(s.y, a.y);
        acc1 += nll_term(s.z, a.z) + nll_term(s.w, a.w);
    }
    // Scalar tail for n not divisible by 4 (no-op for B = 2^24).
    const int tail = n4 << 2;
    for (int i = tail + tid; i < n; i += stride)
        acc0 += nll_term(syn[i], ant[i]);

    // ---- wave32 reduction via V_WMMA_F32_16X16X4_F32 ----
    // A (16x4 f32) layout: VGPR0 = lanes0-15:K0 / lanes16-31:K2,
    //                      VGPR1 = lanes0-15:K1 / lanes16-31:K3
    // -> (acc0, acc1) per lane covers all 64 A elements exactly once.
    // B = ones(4x16)  =>  D[m][n] = sum_k A[m][k]  (row sums, replicated per col).
    v2f amat; amat.x = acc0; amat.y = acc1;
    v2f bmat; bmat.x = 1.0f; bmat.y = 1.0f;
    v8f c = {};
    c = __builtin_amdgcn_wmma_f32_16x16x4_f32(
            /*neg_a=*/false, amat, /*neg_b=*/false, bmat,
            /*c_mod=*/(short)0, c, /*reuse_a=*/false, /*reuse_b=*/false);
    // lanes 0-15 hold rows 0-7 in c[0..7]; lanes 16-31 hold rows 8-15.
    float half = ((c[0] + c[1]) + (c[2] + c[3])) + ((c[4] + c[5]) + (c[6] + c[7]));
    float wsum = half + __shfl_xor(half, 16, 32);   // combine the two half-waves

    if ((threadIdx.x & 31) == 0)
        ws[blockIdx.x * NWAVE + (threadIdx.x >> 5)] = wsum;
}

__global__ __launch_bounds__(256) void loss_final_kernel(
    const float* __restrict__ ws, float* __restrict__ out, int nparts, float inv2B) {
    __shared__ float sm[256];
    float acc = 0.0f;
    for (int i = threadIdx.x; i < nparts; i += 256) acc += ws[i];
    sm[threadIdx.x] = acc;
    __syncthreads();
    for (int s = 128; s > 0; s >>= 1) {
        if ((int)threadIdx.x < s) sm[threadIdx.x] += sm[threadIdx.x + s];
        __syncthreads();
    }
    if (threadIdx.x == 0) out[0] = sm[0] * inv2B;
}

extern "C" void kernel_launch(void* const* d_in, const int* in_sizes, int n_in,
                              void* d_out, int out_size, void* d_ws, size_t ws_size,
                              hipStream_t stream) {
    // d_in[0] = S1_out (unused, only supplies B), d_in[1] = synonymy, d_in[2] = antonymy
    const float* syn = (const float*)d_in[1];
    const float* ant = (const float*)d_in[2];
    const int n = in_sizes[1];
    float* ws = (float*)d_ws;   // NPART floats = 32 KiB

    loss_partial_kernel<<<NBLK, NTHR, 0, stream>>>(syn, ant, ws, n);
    loss_final_kernel<<<1, 256, 0, stream>>>(ws, (float*)d_out, NPART, 0.5f / (float)n);
}